// AdaptiveCNN_1709396984047
// MI455X (gfx1250) — compile-verified
//
#include <hip/hip_runtime.h>
#include <hip/hip_bf16.h>

// ---------------------------------------------------------------------------
// AdaptiveCNN on gfx1250 (wave32):
//   - hypernetwork FCs and per-sample convs as WMMA f32 16x16x4 implicit GEMMs
//   - shared A-tiles staged to LDS (async global->LDS when available)
//   - conv gather offsets folded to compile-time constants (full K unroll)
//   - fused bias + 2x2 maxpool (cross-lane shfl_xor) + branchless SELU
// ---------------------------------------------------------------------------

typedef float v2f __attribute__((ext_vector_type(2)));
typedef float v8f __attribute__((ext_vector_type(8)));
typedef int   v4i __attribute__((ext_vector_type(4)));

// Branchless SELU: lam*max(x,0) + lam*alp*(exp(min(x,0)) - 1).
// x>0  -> exp(0)-1 = 0;  x<=0 -> max term 0. No EXEC-mask divergence.
__device__ __forceinline__ float selu_f(float x) {
    const float lam    = 1.0507009873554805f;
    const float lamalp = 1.0507009873554805f * 1.6732632423543772f;
    return fmaf(lam, fmaxf(x, 0.0f),
                lamalp * (__expf(fminf(x, 0.0f)) - 1.0f));
}

__device__ __forceinline__ v8f wmma_f32(v2f a, v2f b, v8f c) {
    // 8 args: (neg_a, A, neg_b, B, c_mod, C, reuse_a, reuse_b)
    return __builtin_amdgcn_wmma_f32_16x16x4_f32(
        false, a, false, b, (short)0, c, false, false);
}

// Stage a contiguous tile of `nfloats` floats (multiple of 4, 16B-aligned on
// both sides) from global into LDS, cooperatively across the 256-thread block.
// Uses the gfx1250 async global->LDS copy when the toolchain exposes it.
__device__ __forceinline__ void stage_tile(float* __restrict__ ldst,
                                           const float* __restrict__ gsrc,
                                           int nfloats) {
    const int tid = threadIdx.x;
#if __has_builtin(__builtin_amdgcn_global_load_async_to_lds_b128) && \
    __has_builtin(__builtin_amdgcn_s_wait_asynccnt)
    for (int i = tid * 4; i < nfloats; i += 256 * 4)
        __builtin_amdgcn_global_load_async_to_lds_b128(
            (__attribute__((address_space(1))) v4i*)(gsrc + i),
            (__attribute__((address_space(3))) v4i*)(ldst + i), 0, 0);
    __builtin_amdgcn_s_wait_asynccnt(0);
#else
    for (int i = tid * 4; i < nfloats; i += 256 * 4)
        *(float4*)(ldst + i) = *(const float4*)(gsrc + i);
#endif
    __syncthreads();
}

// ---------------------------------------------------------------------------
// FC layer: C[M,N] = selu(A[M,K] @ W[K,N] + bias[N]); row-major.
// Block = 8 waves sharing one 16-row A tile (staged in LDS); one wave per
// 16x16 output tile. K is a template constant (64 or 256).
// ---------------------------------------------------------------------------
template <int K>
__global__ __launch_bounds__(256)
void fc_selu_wmma(const float* __restrict__ A, const float* __restrict__ W,
                  const float* __restrict__ bias, float* __restrict__ C,
                  int N) {
    __shared__ __align__(16) float la[16 * K];
    const int m0 = blockIdx.y << 4;
    stage_tile(la, A + (size_t)m0 * K, 16 * K);   // contiguous 16 rows

    const int ntile = blockIdx.x * 8 + (threadIdx.x >> 5);
    if (ntile >= (N >> 4)) return;                // uniform per-wave branch
    const int lane  = threadIdx.x & 31;
    const int c16   = lane & 15;                  // A-row / B-col / D-col
    const int khalf = lane >> 4;
    const int n0    = ntile << 4;

    const float* __restrict__ Wp = W + n0 + c16;  // this lane's B column
    v8f acc = {};
#pragma unroll 4
    for (int k0 = 0; k0 < K; k0 += 4) {
        const int ka = k0 + khalf * 2;
        v2f a = *(const v2f*)(la + c16 * K + ka); // ds_load_b64
        v2f b;
        b.x = Wp[(size_t)(ka + 0) * N];
        b.y = Wp[(size_t)(ka + 1) * N];
        __builtin_prefetch(Wp + (size_t)(ka + 32) * N, 0, 3); // near-cache stream
        acc = wmma_f32(a, b, acc);
    }

    const float bv = bias[n0 + c16];
#pragma unroll
    for (int r = 0; r < 8; ++r) {
        const int mm = m0 + r + (khalf << 3);     // D: VGPR r -> row r / r+8
        C[(size_t)mm * N + n0 + c16] = selu_f(acc[r] + bv);
    }
}

// ---------------------------------------------------------------------------
// Per-sample conv (implicit GEMM) + bias + optional fused 2x2 maxpool + SELU.
//   D[c,p] = sum_k wb[s][c*CIN*9 + k] * patch[k,p],  pad value = -1.0
// Weight tile (16 x KDIM, contiguous) staged in LDS, shared by all 8 waves.
// K-loop fully unrolled so the (ci,ky,kx) decode folds to constants; the only
// runtime part is a khalf-select between two immediate offsets.
// ---------------------------------------------------------------------------
template <int CIN, int COUT, int H, int W, int PSIZE, bool POOL>
__global__ __launch_bounds__(256)
void conv_wmma(const float* __restrict__ in, const float* __restrict__ wb,
               float* __restrict__ out) {
    constexpr int KDIM   = CIN * 9;
    constexpr int NTILES = POOL ? (H / 2) * (W / 8) : (H * W) / 16;
    static_assert(NTILES % 8 == 0, "all waves active");

    __shared__ __align__(16) float lw[16 * KDIM];
    const int s  = blockIdx.z;
    const int m0 = blockIdx.y << 4;
    const float* __restrict__ wbp = wb + (size_t)s * PSIZE;
    stage_tile(lw, wbp + (size_t)m0 * KDIM, 16 * KDIM);

    const int ntile = blockIdx.x * 8 + (threadIdx.x >> 5);
    const int lane  = threadIdx.x & 31;
    const int c16   = lane & 15;
    const int khalf = lane >> 4;

    int oy, ox;                                    // this lane's D column
    if (POOL) {
        const int ty = ntile / (W / 8);
        const int tx = ntile % (W / 8);
        oy = ty * 2 + (c16 >> 3);
        ox = tx * 8 + (c16 & 7);
    } else {
        const int p = ntile * 16 + c16;
        oy = p / W;
        ox = p % W;
    }

    const float* __restrict__ x    = in + (size_t)s * CIN * H * W;
    const float* __restrict__ base = x + (oy - 1) * W + (ox - 1);

    v8f acc = {};
#pragma unroll
    for (int k0 = 0; k0 < KDIM; k0 += 4) {
        v2f a = *(const v2f*)(lw + c16 * KDIM + k0 + khalf * 2); // ds_load_b64
        v2f b;
#pragma unroll
        for (int t = 0; t < 2; ++t) {
            const int klo = k0 + t, khi = k0 + 2 + t;   // constants after unroll
            const int rlo = klo % 9, rhi = khi % 9;
            const int off_lo = (klo / 9) * H * W + (rlo / 3) * W + (rlo % 3);
            const int off_hi = (khi / 9) * H * W + (rhi / 3) * W + (rhi % 3);
            const int off = khalf ? off_hi : off_lo;     // v_cndmask
            const int ky  = khalf ? (rhi / 3) : (rlo / 3);
            const int kx  = khalf ? (rhi % 3) : (rlo % 3);
            const int iy = oy + ky - 1, ix = ox + kx - 1;
            const float v = ((unsigned)iy < (unsigned)H &&
                             (unsigned)ix < (unsigned)W) ? base[off] : -1.0f;
            if (t == 0) b.x = v; else b.y = v;
        }
        acc = wmma_f32(a, b, acc);
    }

    const float* __restrict__ bias = wbp + (size_t)COUT * KDIM;
#pragma unroll
    for (int r = 0; r < 8; ++r) {
        const int ch = m0 + r + (khalf << 3);       // channel of this D VGPR
        float v = acc[r] + bias[ch];
        if (POOL) {
            const float vx  = __shfl_xor(v, 1);     // x-neighbor
            const float vy  = __shfl_xor(v, 8);     // y-neighbor
            const float vxy = __shfl_xor(v, 9);
            float m = fmaxf(fmaxf(v, vx), fmaxf(vy, vxy));
            m = selu_f(m);
            if ((c16 & 1) == 0 && c16 < 8) {
                out[(((size_t)s * COUT + ch) * (H / 2) + (oy >> 1)) * (W / 2)
                    + (ox >> 1)] = m;
            }
        } else {
            out[(((size_t)s * COUT + ch) * H + oy) * W + ox] = selu_f(v);
        }
    }
}

// ---------------------------------------------------------------------------
// Launcher
// ---------------------------------------------------------------------------
template <int K>
static inline void launch_fc(const float* A, const float* W, const float* b,
                             float* C, int N, hipStream_t stream) {
    const int ntiles = N / 16;
    dim3 grid((ntiles + 7) / 8, 512 / 16, 1);
    fc_selu_wmma<K><<<grid, 256, 0, stream>>>(A, W, b, C, N);
}

extern "C" void kernel_launch(void* const* d_in, const int* in_sizes, int n_in,
                              void* d_out, int out_size, void* d_ws, size_t ws_size,
                              hipStream_t stream) {
    const float* main_in = (const float*)d_in[0];   // (512, 8, 64, 64)
    const float* side    = (const float*)d_in[1];   // (512, 64)

    // workspace layout (floats)
    float* hidA = (float*)d_ws;                       // 512*256
    float* hidB = hidA + (size_t)512 * 256;           // 512*256
    float* wb   = hidB + (size_t)512 * 256;           // 512*9248 (max psize)
    float* act1 = wb   + (size_t)512 * 9248;          // 512*16*32*32
    float* act2 = act1 + (size_t)512 * 16 * 32 * 32;  // 512*32*16*16
    float* out  = (float*)d_out;                      // 512*32*16*16

    const int psizes[3] = {8 * 16 * 9 + 16, 16 * 32 * 9 + 32, 32 * 32 * 9 + 32};

    for (int i = 0; i < 3; ++i) {
        const float* w0 = (const float*)d_in[2 + 6 * i];
        const float* b0 = (const float*)d_in[3 + 6 * i];
        const float* w1 = (const float*)d_in[4 + 6 * i];
        const float* b1 = (const float*)d_in[5 + 6 * i];
        const float* w2 = (const float*)d_in[6 + 6 * i];
        const float* b2 = (const float*)d_in[7 + 6 * i];

        launch_fc<64>(side, w0, b0, hidA, 256, stream);
        launch_fc<256>(hidA, w1, b1, hidB, 256, stream);
        launch_fc<256>(hidB, w2, b2, wb, psizes[i], stream);

        if (i == 0) {
            conv_wmma<8, 16, 64, 64, 8 * 16 * 9 + 16, true>
                <<<dim3(256 / 8, 1, 512), 256, 0, stream>>>(main_in, wb, act1);
        } else if (i == 1) {
            conv_wmma<16, 32, 32, 32, 16 * 32 * 9 + 32, true>
                <<<dim3(64 / 8, 2, 512), 256, 0, stream>>>(act1, wb, act2);
        } else {
            conv_wmma<32, 32, 16, 16, 32 * 32 * 9 + 32, false>
                <<<dim3(16 / 8, 2, 512), 256, 0, stream>>>(act2, wb, out);
        }
    }
}